// MarginLossForSimilarity_91336774517098
// MI455X (gfx1250) — compile-verified
//
#include <hip/hip_runtime.h>

typedef __attribute__((ext_vector_type(2))) float v2f;
typedef __attribute__((ext_vector_type(4))) float v4f;
typedef __attribute__((ext_vector_type(8))) float v8f;

#define ALPHA_C   0.3f
#define BETHE_C   1.2f
#define K_DIM     128
#define TILE      128
#define LDS_STRIDE 132   // 128 + 4 pad: rows stay 16B-aligned, row stride shifts banks by 4

#if __has_builtin(__builtin_amdgcn_sched_barrier)
#define SCHED_FENCE() __builtin_amdgcn_sched_barrier(0)
#else
#define SCHED_FENCE()
#endif

__global__ void ml_zero_out_kernel(float* out) { out[0] = 0.0f; }

__global__ __launch_bounds__(256)
void margin_loss_wmma_f32(const float* __restrict__ X,
                          const long long* __restrict__ labels,
                          float* __restrict__ out,
                          float inv_n) {
    extern __shared__ float smem[];
    float* Xa = smem;                       // [TILE][LDS_STRIDE] rows i0..i0+127
    float* Xb = smem + TILE * LDS_STRIDE;   // [TILE][LDS_STRIDE] rows j0..j0+127
    __shared__ float red[8];

    // ---- triangular tile decode: block t -> (r, c) with c <= r ----
    int t = blockIdx.x;
    int r = 0;
    while ((r + 1) * (r + 2) / 2 <= t) ++r;     // SALU-only, <=64 iters
    int c = t - r * (r + 1) / 2;
    const int i0 = r * TILE;      // M-side rows
    const int j0 = c * TILE;      // N-side rows
    const float weight = (r == c) ? 1.0f : 2.0f;

    const int tid = threadIdx.x;

    // ---- cooperative global -> LDS staging (float4 per lane, 16 chunks) ----
    #pragma unroll
    for (int chunk = 0; chunk < 16; ++chunk) {
        int idx  = chunk * 256 + tid;        // 0..4095  (128 rows * 32 float4)
        int row  = idx >> 5;
        int col4 = (idx & 31) << 2;
        v4f a = *(const v4f*)(X + (size_t)(i0 + row) * K_DIM + col4);
        v4f b = *(const v4f*)(X + (size_t)(j0 + row) * K_DIM + col4);
        *(v4f*)&Xa[row * LDS_STRIDE + col4] = a;
        *(v4f*)&Xb[row * LDS_STRIDE + col4] = b;
    }
    __syncthreads();

    const int wave  = tid >> 5;      // 8 waves, wave32
    const int lane  = tid & 31;
    const int lhalf = lane >> 4;     // 0: K=0..1 of quad, 1: K=2..3 of quad
    const int l15   = lane & 15;

    // ---- A fragments for the whole K=128 strip (wave covers M rows m0..m0+15)
    // ISA A 16x4 layout: lanes 0-15 hold M=lane (K=0,1), lanes 16-31 M=lane-16 (K=2,3)
    const int m0 = wave * 16;
    v2f afrag[32];
    #pragma unroll
    for (int ks = 0; ks < 32; ++ks) {
        int kk = ks * 4 + lhalf * 2;
        afrag[ks] = *(const v2f*)&Xa[(m0 + l15) * LDS_STRIDE + kk];
    }

    // labels for the M rows this lane's C elements map to:
    // C layout: VGPR rr -> M = rr (lanes 0-15) / M = rr+8 (lanes 16-31)
    int labi[8];
    #pragma unroll
    for (int rr = 0; rr < 8; ++rr)
        labi[rr] = (int)labels[i0 + m0 + rr + lhalf * 8];

    float local = 0.0f;
    for (int nsub = 0; nsub < 8; ++nsub) {
        const int n0 = nsub * 16;

        // ---- fetch the whole B panel for this subtile first (16 ds_load_2addr_b64),
        //      so the WMMA chain below runs with a single s_wait_dscnt ----
        v2f bfrag[32];
        const float* bbase = &Xb[(n0 + l15) * LDS_STRIDE + lhalf * 2];
        SCHED_FENCE();
        #pragma unroll
        for (int ks = 0; ks < 32; ++ks)
            bfrag[ks] = *(const v2f*)(bbase + ks * 4);
        SCHED_FENCE();

        v8f acc = {};
        #pragma unroll
        for (int ks = 0; ks < 32; ++ks) {
            acc = __builtin_amdgcn_wmma_f32_16x16x4_f32(
                      /*neg_a=*/false, afrag[ks],
                      /*neg_b=*/false, bfrag[ks],
                      /*c_mod=*/(short)0, acc,
                      /*reuse_a=*/false, /*reuse_b=*/false);
        }
        SCHED_FENCE();

        int labj = (int)labels[j0 + n0 + l15];
        #pragma unroll
        for (int rr = 0; rr < 8; ++rr) {
            float s = (labi[rr] == labj) ? 1.0f : -1.0f;
            local += fmaxf(ALPHA_C + s * (acc[rr] - BETHE_C), 0.0f);
        }
    }

    // ---- wave32 shuffle reduction, then block reduce, one atomic per block ----
    #pragma unroll
    for (int off = 16; off > 0; off >>= 1)
        local += __shfl_down(local, off, 32);
    if (lane == 0) red[wave] = local;
    __syncthreads();
    if (tid == 0) {
        float s = 0.0f;
        #pragma unroll
        for (int w = 0; w < 8; ++w) s += red[w];
        atomicAdd(out, s * weight * inv_n);
    }
}

extern "C" void kernel_launch(void* const* d_in, const int* in_sizes, int n_in,
                              void* d_out, int out_size, void* d_ws, size_t ws_size,
                              hipStream_t stream) {
    const float*     X      = (const float*)d_in[0];
    const long long* labels = (const long long*)d_in[1];
    float*           out    = (float*)d_out;

    const int N  = in_sizes[1];          // 8192 labels
    const int NB = N / TILE;             // 64 tiles per side
    const int nblocks = NB * (NB + 1) / 2;   // lower-triangular tiles only

    ml_zero_out_kernel<<<1, 1, 0, stream>>>(out);

    const size_t lds_bytes = (size_t)2 * TILE * LDS_STRIDE * sizeof(float); // ~132 KB
    margin_loss_wmma_f32<<<nblocks, 256, lds_bytes, stream>>>(
        X, labels, out, 1.0f / (float)N);
}